// MinGRU2_43602507989236
// MI455X (gfx1250) — compile-verified
//
#include <hip/hip_runtime.h>

#define T_LEN 4096
#define NTHR  256
#define PER   16        // elements per thread: 256*16 = 4096
#define EPS_V 1e-6f
#define PADSZ 5120      // 4096 + 4 floats of pad per 16-float chunk

typedef int v4i __attribute__((ext_vector_type(4)));
typedef __attribute__((address_space(1))) v4i as1_v4i;
typedef __attribute__((address_space(3))) v4i as3_v4i;
typedef __attribute__((address_space(3))) float as3_float;

// padded LDS index: +4 floats (16B) every 16 floats -> lane stride 20 floats
// (2-way bank conflicts max instead of 8-way), 16B alignment preserved
__device__ __forceinline__ int padi(int e) { return e + ((e >> 4) << 2); }

// ---- CDNA5 async global->LDS copy (16B per lane, ASYNCcnt-tracked) ----
__device__ __forceinline__ void async_copy_b128(const float* gptr, float* lptr) {
#if defined(__HIP_DEVICE_COMPILE__)
#if __has_builtin(__builtin_amdgcn_global_load_async_to_lds_b128)
  __builtin_amdgcn_global_load_async_to_lds_b128(
      (as1_v4i*)gptr, (as3_v4i*)lptr, /*offset=*/0, /*cpol=*/0);
#else
  unsigned lds_addr = (unsigned)(unsigned long long)(as3_float*)lptr;
  unsigned long long ga = (unsigned long long)gptr;
  asm volatile("global_load_async_to_lds_b128 %0, %1, off"
               :: "v"(lds_addr), "v"(ga) : "memory");
#endif
#else
  (void)gptr; (void)lptr;
#endif
}

__device__ __forceinline__ void wait_async0() {
#if defined(__HIP_DEVICE_COMPILE__)
#if __has_builtin(__builtin_amdgcn_s_wait_asynccnt)
  __builtin_amdgcn_s_wait_asynccnt(0);
#else
  asm volatile("s_wait_asynccnt 0" ::: "memory");
#endif
#endif
}

// One workgroup per (batch, out-channel) row: chunked parallel scan of
// H_t = a_t * H_{t-1} + b_t over T=4096, a=sigmoid(-g), b=sigmoid(g)*sgn(h)*max(|h|,eps)
__global__ __launch_bounds__(NTHR) void mingru_scan_kernel(const float* __restrict__ x,
                                                           float* __restrict__ out) {
  __shared__ __align__(16) float sh[PADSZ];   // h row (padded), reused as output stage
  __shared__ __align__(16) float sg[PADSZ];   // g row (padded)
  __shared__ float wA[8], wB[8], wBp[8];

  const int tid = threadIdx.x;
  const int r   = blockIdx.x;          // 0 .. 4095
  const int b   = r >> 8;              // batch
  const int c   = r & 255;             // output channel
  const bool bwd = (c >= 128);
  const int cc  = bwd ? (c - 128) : c;
  const int hch = bwd ? (256 + cc) : cc;
  const int gch = bwd ? (384 + cc) : (cc + 128);

  const float* hrow = x   + ((size_t)b * 512 + hch) * T_LEN;
  const float* grow = x   + ((size_t)b * 512 + gch) * T_LEN;
  float*       orow = out + ((size_t)b * 256 + c)   * T_LEN;

  // ---- stage both 16KB rows into padded LDS via async b128 copies ----
  // e % 16 in {0,4,8,12}: each 16B transfer stays inside one pad group, aligned.
  #pragma unroll
  for (int i = 0; i < 4; ++i) {
    const int e = (i * NTHR + tid) * 4;           // element offset, 16B per lane
    async_copy_b128(hrow + e, sh + padi(e));
    async_copy_b128(grow + e, sg + padi(e));
  }
  wait_async0();
  __syncthreads();

  // ---- pass 1: per-element coefficients + per-thread chunk aggregate ----
  const int p0 = tid * PER;                       // scan position of first element
  float av[PER], bv[PER];
  float A = 1.0f, B = 0.0f;                       // composed transform of chunk
  #pragma unroll
  for (int j = 0; j < 4; ++j) {
    // forward: group j covers t = p0+4j .. p0+4j+3 (in order)
    // backward: group j covers t = 4092-p0-4j .. 4095-p0-4j (consumed reversed)
    const int tb = bwd ? (T_LEN - 4 - p0 - 4 * j) : (p0 + 4 * j);
    float h4[4], g4[4];
    *(float4*)h4 = *(const float4*)(sh + padi(tb));
    *(float4*)g4 = *(const float4*)(sg + padi(tb));
    #pragma unroll
    for (int m = 0; m < 4; ++m) {
      const int k = 4 * j + m;
      const int cmp = bwd ? (3 - m) : m;          // component in scan order
      const float h = h4[cmp];
      const float g = g4[cmp];
      const float e   = __expf(-g);
      const float sg_ = __builtin_amdgcn_rcpf(1.0f + e); // sigmoid(g)
      const float a   = e * sg_;                         // sigmoid(-g)
      const float bb  = copysignf(fmaxf(fabsf(h), EPS_V), h) * sg_;
      av[k] = a; bv[k] = bb;
      B = fmaf(a, B, bb);
      A *= a;
    }
  }

  // ---- wave32 inclusive scan of (A,B) via shuffles ----
  const int lane = tid & 31;
  const int wid  = tid >> 5;
  float Ai = A, Bi = B;
  #pragma unroll
  for (int d = 1; d < 32; d <<= 1) {
    const float Au = __shfl_up(Ai, (unsigned)d, 32);
    const float Bu = __shfl_up(Bi, (unsigned)d, 32);
    if (lane >= d) { Bi = fmaf(Ai, Bu, Bi); Ai *= Au; }
  }
  // exclusive-within-wave prefix
  float Ax = __shfl_up(Ai, 1u, 32);
  float Bx = __shfl_up(Bi, 1u, 32);
  if (lane == 0) { Ax = 1.0f; Bx = 0.0f; }
  if (lane == 31) { wA[wid] = Ai; wB[wid] = Bi; }   // wave totals
  __syncthreads();

  // ---- combine 8 wave aggregates (exclusive prefixes) ----
  if (tid == 0) {
    float RB = 0.0f;
    #pragma unroll
    for (int w = 0; w < 8; ++w) {
      wBp[w] = RB;
      RB = fmaf(wA[w], RB, wB[w]);
    }
  }
  __syncthreads();

  // starting state for this thread's chunk (H_0 = 0 => value = composed B)
  float H = fmaf(Ax, wBp[wid], Bx);

  // ---- pass 2: replay chunk from registers, store at original time index ----
  #pragma unroll
  for (int j = 0; j < 4; ++j) {
    const int tb = bwd ? (T_LEN - 4 - p0 - 4 * j) : (p0 + 4 * j);
    float o4[4];
    #pragma unroll
    for (int m = 0; m < 4; ++m) {
      const int k = 4 * j + m;
      H = fmaf(av[k], H, bv[k]);
      o4[bwd ? (3 - m) : m] = H;
    }
    *(float4*)(sh + padi(tb)) = *(const float4*)o4;   // reuse sh as output stage
  }
  __syncthreads();

  // ---- coalesced float4 writeback (LDS read is aligned, within pad group) ----
  #pragma unroll
  for (int i = 0; i < 4; ++i) {
    const int e = (i * NTHR + tid) * 4;
    *(float4*)(orow + e) = *(const float4*)(sh + padi(e));
  }
}

extern "C" void kernel_launch(void* const* d_in, const int* in_sizes, int n_in,
                              void* d_out, int out_size, void* d_ws, size_t ws_size,
                              hipStream_t stream) {
  (void)in_sizes; (void)n_in; (void)out_size; (void)d_ws; (void)ws_size;
  const float* x = (const float*)d_in[0];
  float* out = (float*)d_out;
  // 16 batches * 256 (channel,direction) rows = 4096 workgroups
  hipLaunchKernelGGL(mingru_scan_kernel, dim3(16 * 256), dim3(NTHR), 0, stream, x, out);
}